// StereoMatching_21912923144675
// MI455X (gfx1250) — compile-verified
//
#include <hip/hip_runtime.h>
#include <hip/hip_bf16.h>
#include <cstddef>

typedef __attribute__((ext_vector_type(2))) float v2f;
typedef __attribute__((ext_vector_type(4))) float v4f;
typedef __attribute__((ext_vector_type(8))) float v8f;

#define BATCH 2
#define Hh    384
#define Ww    1248
#define DISP  192
#define WT    (Ww / 16)    // 78 w-tiles
#define HT    (Hh / 16)    // 24 h-tiles
#define WGW   6            // w-tiles per block (6 waves, 192 threads)
#define WQ    (WT / WGW)   // 13 w-groups
#define NBLK  (BATCH * HT * WQ)  // 624

#define SLAB_COLS   288    // 192 halo + 6*16 tile columns
#define SLAB_STRIDE 292    // pad: 36*row mod 64 covers all multiples of 4 -> conflict-free b64 reads

// out[b,h,w] = l[b,h,w] * sum_{d=0..191, w-d>=0} d * r[b,h,w-d]
// Per 16x16 tile: D = A(16x208) x Bband(208x16) via 52 x V_WMMA_F32_16X16X4_F32.
// A slab staged through LDS (coalesced b128 fill, 4.3x reuse across 6 waves),
// Bband generated in registers with a max() recurrence.
__global__ __launch_bounds__(192) void StereoMatching_wmma_kernel(
    const float* __restrict__ L, const float* __restrict__ R, float* __restrict__ out)
{
    __shared__ __align__(16) float slab[16 * SLAB_STRIDE];   // 18688 B

    const int tid  = threadIdx.x;
    const int lane = tid & 31;
    const int widx = tid >> 5;          // wave -> w-tile within group

    const int wq = blockIdx.x % WQ;
    const int t2 = blockIdx.x / WQ;
    const int ht = t2 % HT;
    const int b  = t2 / HT;
    const int h0 = ht << 4;
    const int gw0 = wq * (WGW * 16) - DISP;   // global col of slab col 0 (may be < 0)

    // ---- cooperative slab fill: 16 rows x 288 cols of r, zero-padded left ----
    {
        const int frow  = tid / 12;          // 16 rows x 12 threads
        const int fcol0 = (tid % 12) * 24;   // 24 contiguous floats per thread
        const float* grow = R + (size_t)(b * Hh + h0 + frow) * Ww;
        float* srow = &slab[frow * SLAB_STRIDE];
        if (gw0 >= 0) {                       // block-uniform branch
#pragma unroll
            for (int j = 0; j < 6; ++j) {     // 6 x b128, 16B aligned both sides
                const int col = fcol0 + 4 * j;
                *(v4f*)(srow + col) = *(const v4f*)(grow + gw0 + col);
            }
        } else {
#pragma unroll
            for (int j = 0; j < 24; ++j) {
                const int col = fcol0 + j;
                const int gc  = gw0 + col;
                const int gcc = gc < 0 ? 0 : gc;
                const float v = grow[gcc];
                srow[col] = (gc >= 0) ? v : 0.0f;
            }
        }
    }
    __syncthreads();

    // ---- WMMA lane roles (ISA 7.12.2, wave32) ----
    //   A 16x4 f32 : lane&15 = M, lane>=16 -> K pair {2,3}; vgpr = K lsb
    //   B 4x16 f32 : lane&15 = N, lane>=16 -> K pair {2,3}; vgpr = K lsb
    //   C/D 16x16  : lane&15 = N, vgpr v -> M = v + 8*(lane>=16)
    const int n     = lane & 15;
    const int khalf = lane >> 4;

    // A row for this lane, pre-offset by tile column and K half.
    const float* arow = &slab[n * SLAB_STRIDE + widx * 16 + 2 * khalf];
    v8f c = {};

    // Peel kc=0..3: kk = 4kc+2khalf <= 14, x0 = n+192-kk in [178,207] -> only upper check.
#pragma unroll
    for (int kc = 0; kc < 4; ++kc) {
        const v2f a = *(const v2f*)(arow + 4 * kc);   // ds_load_b64, 8B aligned
        const int x0 = n + DISP - (4 * kc + 2 * khalf);
        v2f bb;
        bb.x = (x0     < DISP) ? (float)x0       : 0.0f;
        bb.y = (x0 - 1 < DISP) ? (float)(x0 - 1) : 0.0f;
        c = __builtin_amdgcn_wmma_f32_16x16x4_f32(false, a, false, bb,
                                                  (short)0, c, false, false);
    }

    // Main kc=4..51: x <= 191 guaranteed; band recurrence f(x-4)=max(f(x)-4,0).
    float xf0 = (float)(n + DISP - 16 - 2 * khalf);
    float xf1 = xf0 - 1.0f;
#pragma unroll 4
    for (int kc = 4; kc < 52; ++kc) {
        const v2f a = *(const v2f*)(arow + 4 * kc);
        v2f bb;
        bb.x = fmaxf(xf0, 0.0f);
        bb.y = fmaxf(xf1, 0.0f);
        c = __builtin_amdgcn_wmma_f32_16x16x4_f32(false, a, false, bb,
                                                  (short)0, c, false, false);
        xf0 -= 4.0f;
        xf1 -= 4.0f;
    }

    // ---- epilogue: out = l * D ----
    const int w0 = (wq * WGW + widx) << 4;
    const size_t obase = (size_t)(b * Hh + h0) * Ww + (size_t)(w0 + n);
#pragma unroll
    for (int v = 0; v < 8; ++v) {
        const int M = v + (khalf << 3);
        const size_t idx = obase + (size_t)M * Ww;
        out[idx] = L[idx] * c[v];
    }
}

extern "C" void kernel_launch(void* const* d_in, const int* in_sizes, int n_in,
                              void* d_out, int out_size, void* d_ws, size_t ws_size,
                              hipStream_t stream) {
    (void)in_sizes; (void)n_in; (void)out_size; (void)d_ws; (void)ws_size;
    const float* L = (const float*)d_in[0];   // left_img
    const float* R = (const float*)d_in[1];   // right_img
    float* O = (float*)d_out;                 // (B,H,W) f32

    static_assert(WT % WGW == 0, "w-tiles must split into groups");
    dim3 grid(NBLK);     // 624
    dim3 block(192);     // 6 waves of 32
    hipLaunchKernelGGL(StereoMatching_wmma_kernel, grid, block, 0, stream, L, R, O);
}